// RNNFreqAndTime_2826088480933
// MI455X (gfx1250) — compile-verified
//
#include <hip/hip_runtime.h>
#include <hip/hip_bf16.h>

// ---------------------------------------------------------------------------
// RNN autoencoder (freq/time) for MI455X gfx1250.
// All GEMMs (input projections and recurrent h@Whh^T) run on the f32 WMMA
// pipe: v_wmma_f32_16x16x4_f32, one 16x16 C tile per wave32.
// Recurrent scans: Whh staged in LDS once (reused for all T steps), grid-wide
// counter barrier between timesteps, h ping-pong in global (L2-resident).
// ---------------------------------------------------------------------------

typedef __attribute__((ext_vector_type(2))) float v2f;
typedef __attribute__((ext_vector_type(8))) float v8f;

#define BATCH 32

// ---------------------------------------------------------------------------
// Grid-wide barrier (monotonic counter, release/acquire fences). All blocks
// of the scan kernel are resident (<=16 blocks), so spinning is safe.
// ---------------------------------------------------------------------------
__device__ __forceinline__ void grid_sync(unsigned* cnt, unsigned nblk, unsigned* step) {
  __builtin_amdgcn_fence(__ATOMIC_RELEASE, "agent");
  __syncthreads();
  if (nblk > 1) {
    if (threadIdx.x == 0) {
      __hip_atomic_fetch_add(cnt, 1u, __ATOMIC_RELEASE, __HIP_MEMORY_SCOPE_AGENT);
      const unsigned target = (*step + 1u) * nblk;
      while (__hip_atomic_load(cnt, __ATOMIC_ACQUIRE, __HIP_MEMORY_SCOPE_AGENT) < target) {
        __builtin_amdgcn_s_sleep(1);
      }
    }
    __syncthreads();
  }
  __builtin_amdgcn_fence(__ATOMIC_ACQUIRE, "agent");
  ++*step;
}

__global__ void zero_u32_kernel(unsigned* p, int n) {
  if ((int)threadIdx.x < n) p[threadIdx.x] = 0u;
}

// ---------------------------------------------------------------------------
// Input projection: XW[M,H] = X[M,I] @ Wih^T + (bih + bhh)
// M = T*BATCH (time-major rows), one 16x16 tile per wave, f32 WMMA K=4 steps.
// f32 WMMA 16x16x4 lane layout (wave32):
//   A (16x4): lane L: m = L&15, a = {A[m][2*(L>>4)], A[m][2*(L>>4)+1]}
//   B (4x16): lane L: n = L&15, b = {B[2*(L>>4)][n], B[2*(L>>4)+1][n]}
//   C/D:      lane L: n = L&15, acc[i] = C[i + 8*(L>>4)][n]
// ---------------------------------------------------------------------------
__global__ void gemm_xw_wmma(const float* __restrict__ X, const float* __restrict__ W,
                             const float* __restrict__ bih, const float* __restrict__ bhh,
                             float* __restrict__ XW, int M, int I, int H) {
  const int lane = threadIdx.x & 31;
  const int l15  = lane & 15;
  const int half = lane >> 4;
  const int wave = (int)((blockIdx.x * blockDim.x + threadIdx.x) >> 5);
  const int Ht = H >> 4;
  const int tiles = (M >> 4) * Ht;
  if (wave >= tiles) return;                     // wave-uniform exit
  const int m0 = (wave / Ht) << 4;
  const int n0 = (wave % Ht) << 4;

  const long arow = (long)(m0 + l15) * I + 2 * half;
  const long brow = (long)(n0 + l15) * I + 2 * half;
  v8f acc = {};
#pragma unroll 4
  for (int k = 0; k < I; k += 4) {
    v2f a = *(const v2f*)(X + arow + k);
    v2f b = *(const v2f*)(W + brow + k);
    acc = __builtin_amdgcn_wmma_f32_16x16x4_f32(false, a, false, b, (short)0, acc,
                                                false, false);
  }
  const float bsum = bih[n0 + l15] + bhh[n0 + l15];
  float* outp = XW + (long)(m0 + (half << 3)) * H + n0 + l15;
#pragma unroll
  for (int i = 0; i < 8; ++i) outp[(long)i * H] = acc[i] + bsum;
}

// ---------------------------------------------------------------------------
// Recurrent scan: for t in [0,T): h = tanh(XW[t] + h @ Whh^T); YS[t] = h.
// XW, YS time-major [T, 32, H]. Whh [H,H] row-major. h0/h1: [32,H] ping-pong.
// Tiling is m-minor: global wave w -> m-tile (w&1), n-tile (w>>1); each block
// stages its 64 (or 32) Whh rows into LDS once, padded pitch H+4 to spread
// the 64 LDS banks.
// ---------------------------------------------------------------------------
__global__ void rnn_scan_wmma(const float* __restrict__ XW, const float* __restrict__ Whh,
                              float* __restrict__ YS, float* __restrict__ h0,
                              float* __restrict__ h1, unsigned* __restrict__ cnt,
                              int T, int H) {
  extern __shared__ float s_whh[];
  const int pitch = H + 4;
  const int lane = threadIdx.x & 31;
  const int l15  = lane & 15;
  const int half = lane >> 4;
  const int wpb  = (int)(blockDim.x >> 5);
  const int wave = (int)(threadIdx.x >> 5);
  const int gwave = (int)blockIdx.x * wpb + wave;
  const int m0 = (gwave & 1) << 4;
  const int n0 = (gwave >> 1) << 4;
  const int nRowsBlk = wpb << 3;                       // (wpb/2)*16 rows per block
  const int rowBase  = (((int)blockIdx.x * wpb) >> 1) << 4;

  // Stage this block's Whh rows into LDS (read once, reused for all T steps).
  for (int r = wave; r < nRowsBlk; r += wpb)
    for (int c = lane; c < H; c += 32)
      s_whh[r * pitch + c] = Whh[(long)(rowBase + r) * H + c];

  // h(-1) = 0
  const int gtid = (int)(blockIdx.x * blockDim.x + threadIdx.x);
  const int nthr = (int)(gridDim.x * blockDim.x);
  for (int idx = gtid; idx < BATCH * H; idx += nthr) h0[idx] = 0.0f;

  unsigned step = 0;
  grid_sync(cnt, gridDim.x, &step);

  const long bline = (long)((n0 - rowBase) + l15) * pitch + 2 * half;
  const long aline = (long)(m0 + l15) * H + 2 * half;
  const int n = n0 + l15;

  for (int t = 0; t < T; ++t) {
    const float* hp = (t & 1) ? h1 : h0;
    float* hc = (t & 1) ? h0 : h1;
    const float* xwt = XW + (long)t * BATCH * H;
    if (t + 1 < T)  // pull next timestep's xw slab toward the WGP
      __builtin_prefetch(XW + (long)(t + 1) * BATCH * H + (long)(m0 + l15) * H + n0, 0, 1);

    v8f acc = {};
#pragma unroll 4
    for (int k = 0; k < H; k += 4) {
      v2f a = *(const v2f*)(hp + aline + k);
      v2f b = *(const v2f*)(s_whh + bline + k);
      acc = __builtin_amdgcn_wmma_f32_16x16x4_f32(false, a, false, b, (short)0, acc,
                                                  false, false);
    }
    float* ysp = YS + (long)t * BATCH * H;
#pragma unroll
    for (int i = 0; i < 8; ++i) {
      const int m = m0 + (half << 3) + i;
      const float v = tanhf(acc[i] + xwt[(long)m * H + n]);
      hc[(long)m * H + n] = v;
      ysp[(long)m * H + n] = v;
    }
    grid_sync(cnt, gridDim.x, &step);
  }
}

// ---------------------------------------------------------------------------
// Batched 2D transpose through LDS. Value in[b*si_b + i*si_i + j] goes to
// out[b*so_b + j*so_j + i]. in is contiguous in j, out contiguous in i.
// All dims are multiples of 32 here. Block: 32x8, tile 32x32.
// ---------------------------------------------------------------------------
__global__ void permute3_swap(const float* __restrict__ in, float* __restrict__ out,
                              long si_b, long si_i, long so_b, long so_j) {
  __shared__ float tile[32][33];
  const int b  = (int)blockIdx.z;
  const int i0 = (int)blockIdx.y * 32;
  const int j0 = (int)blockIdx.x * 32;
  const int tx = (int)threadIdx.x;           // 0..31
  const int ty = (int)threadIdx.y;           // 0..7
  const float* inb = in + (long)b * si_b;
#pragma unroll
  for (int r = 0; r < 32; r += 8)
    tile[ty + r][tx] = inb[(long)(i0 + ty + r) * si_i + (j0 + tx)];
  __syncthreads();
  float* outb = out + (long)b * so_b;
#pragma unroll
  for (int r = 0; r < 32; r += 8)
    outb[(long)(j0 + ty + r) * so_j + (i0 + tx)] = tile[tx][ty + r];
}

// ---------------------------------------------------------------------------
extern "C" void kernel_launch(void* const* d_in, const int* in_sizes, int n_in,
                              void* d_out, int out_size, void* d_ws, size_t ws_size,
                              hipStream_t stream) {
  (void)in_sizes; (void)n_in; (void)out_size; (void)ws_size;
  // Input order: x, enc_freq(l0 wih,whh,bih,bhh; l1 ...), dec_freq, enc_time, dec_time
  const float* x = (const float*)d_in[0];
  auto P = [&](int i) { return (const float*)d_in[i]; };
  const int EF = 1, DF = 9, ET = 17, DT = 25;  // base index of each RNN stack

  float* B0 = (float*)d_ws;                           // 64 MB
  float* B1 = B0 + (size_t)16 * 1024 * 1024;          // 64 MB
  float* Ha = B1 + (size_t)16 * 1024 * 1024;          // 128 KB  (h ping)
  float* Hb = Ha + 32 * 1024;                         // 128 KB  (h pong)
  unsigned* cnts = (unsigned*)(Hb + 32 * 1024);       // 8 barrier counters

  // Allow 263 KB dynamic LDS for the H=1024 scan (320 KB/WGP on CDNA5).
  (void)hipFuncSetAttribute((const void*)rnn_scan_wmma,
                            hipFuncAttributeMaxDynamicSharedMemorySize, 263168);

  zero_u32_kernel<<<1, 32, 0, stream>>>(cnts, 8);

  auto gemm = [&](const float* X, int base, int layer, float* XW, int M, int I, int H) {
    const float* wih = P(base + 4 * layer + 0);
    const float* bih = P(base + 4 * layer + 2);
    const float* bhh = P(base + 4 * layer + 3);
    const int tiles = (M / 16) * (H / 16);
    gemm_xw_wmma<<<(tiles + 7) / 8, 256, 0, stream>>>(X, wih, bih, bhh, XW, M, I, H);
  };
  int cid = 0;
  auto scan = [&](const float* XW, int base, int layer, float* YS, int T, int H) {
    const float* whh = P(base + 4 * layer + 1);
    const int tiles = 2 * (H / 16);
    const int wpb = (tiles >= 8) ? 8 : tiles;
    const int blocks = tiles / wpb;
    const size_t lds = (size_t)(wpb * 8) * (size_t)(H + 4) * sizeof(float);
    rnn_scan_wmma<<<blocks, wpb * 32, lds, stream>>>(XW, whh, YS, Ha, Hb,
                                                     cnts + cid++, T, H);
  };

  // ---- stage 1: enc_freq (T=512, 1024 -> 64 -> 64) ----------------------
  // x [32][1024][512] -> B0 time-major [512][32][1024]
  permute3_swap<<<dim3(512 / 32, 1024 / 32, BATCH), dim3(32, 8), 0, stream>>>(
      x, B0, (long)1024 * 512, 512, 1024, (long)BATCH * 1024);
  gemm(B0, EF, 0, B1, 512 * BATCH, 1024, 64);
  scan(B1, EF, 0, B0, 512, 64);
  gemm(B0, EF, 1, B1, 512 * BATCH, 64, 64);
  scan(B1, EF, 1, B0, 512, 64);                 // ys1 [512][32][64] in B0

  // ---- stage 2: enc_time (T=64, 512 -> 32 -> 32) ------------------------
  // [512][32][64] -> [64][32][512]
  permute3_swap<<<dim3(64 / 32, 512 / 32, BATCH), dim3(32, 8), 0, stream>>>(
      B0, B1, 64, (long)BATCH * 64, 512, (long)BATCH * 512);
  gemm(B1, ET, 0, B0, 64 * BATCH, 512, 32);
  scan(B0, ET, 0, B1, 64, 32);
  gemm(B1, ET, 1, B0, 64 * BATCH, 32, 32);
  scan(B0, ET, 1, B1, 64, 32);                  // ys2 [64][32][32] in B1

  // ---- stage 3: dec_time (T=64, 32 -> 512 -> 512); input already time-major
  gemm(B1, DT, 0, B0, 64 * BATCH, 32, 512);
  scan(B0, DT, 0, B1, 64, 512);
  gemm(B1, DT, 1, B0, 64 * BATCH, 512, 512);
  scan(B0, DT, 1, B1, 64, 512);                 // ys3 [64][32][512] in B1

  // ---- stage 4: dec_freq (T=512, 64 -> 1024 -> 1024) --------------------
  // [64][32][512] -> [512][32][64]
  permute3_swap<<<dim3(512 / 32, 64 / 32, BATCH), dim3(32, 8), 0, stream>>>(
      B1, B0, 512, (long)BATCH * 512, 64, (long)BATCH * 64);
  gemm(B0, DF, 0, B1, 512 * BATCH, 64, 1024);
  scan(B1, DF, 0, B0, 512, 1024);
  gemm(B0, DF, 1, B1, 512 * BATCH, 1024, 1024);
  scan(B1, DF, 1, B0, 512, 1024);               // ys4 [512][32][1024] in B0

  // ---- final: [512][32][1024] -> out [32][1024][512] --------------------
  permute3_swap<<<dim3(1024 / 32, 512 / 32, BATCH), dim3(32, 8), 0, stream>>>(
      B0, (float*)d_out, 1024, (long)BATCH * 1024, (long)1024 * 512, 512);
}